// FlaxHouseholderRoPE_78443282694777
// MI455X (gfx1250) — compile-verified
//
#include <hip/hip_runtime.h>

typedef float v2f   __attribute__((ext_vector_type(2)));
typedef float v4f   __attribute__((ext_vector_type(4)));
typedef float v8f   __attribute__((ext_vector_type(8)));
typedef int   v4i_t __attribute__((vector_size(16)));   // async builtin param type

// Problem dims (fixed by reference)
#define B_ 2
#define S_ 4096
#define H_ 32
#define D_ 128
#define R_ 2
static constexpr int   SB        = S_ / 16;             // 256 s-blocks
static constexpr int   BSH       = B_ * S_ * H_;        // rows per tensor
static constexpr int   GPT       = B_ * SB * H_;        // 16384 groups per tensor
static constexpr int   NG        = 2 * GPT;             // 32768 groups (q then k)
static constexpr int   ROWSTRIDE = H_ * D_;             // 4096 elems between s, s+1
static constexpr float EPS_      = 1e-6f;
static constexpr float LOG2_BASE = 13.28771237954945f;  // log2(10000)

static constexpr int WPB  = 2;          // waves per block (wave32)
static constexpr int BLK  = WPB * 32;   // 64 threads
static constexpr int PIPE = 2;          // tile double-buffer
static constexpr int PADD = 132;        // padded row => conflict-free fragments

// ---------- CDNA5 async global->LDS (per-lane 16B), ASYNCcnt-tracked ----------
#if __has_builtin(__builtin_amdgcn_global_load_async_to_lds_b128)
#define HAVE_ASYNC_BUILTIN 1
#endif

__device__ __forceinline__ void async_b128(const float* g, float* lds) {
#if defined(HAVE_ASYNC_BUILTIN)
  __builtin_amdgcn_global_load_async_to_lds_b128(
      (__attribute__((address_space(1))) v4i_t*)(unsigned long long)g,
      (__attribute__((address_space(3))) v4i_t*)(unsigned)(unsigned long long)lds,
      0, 0);
#else
  unsigned           l  = (unsigned)(unsigned long long)lds;
  unsigned long long ga = (unsigned long long)g;
  asm volatile("global_load_async_to_lds_b128 %0, %1, off"
               :: "v"(l), "v"(ga) : "memory");
#endif
}

template <int N>
__device__ __forceinline__ void wait_asynccnt() {
#if __has_builtin(__builtin_amdgcn_s_wait_asynccnt)
  __builtin_amdgcn_s_wait_asynccnt(N);
#else
  asm volatile("s_wait_asynccnt %0" :: "i"(N) : "memory");
#endif
}

__device__ __forceinline__ void wait_dscnt0() {
#if __has_builtin(__builtin_amdgcn_s_wait_dscnt)
  __builtin_amdgcn_s_wait_dscnt(0);
#else
  asm volatile("s_wait_dscnt 0" ::: "memory");
#endif
}

#if __has_builtin(__builtin_amdgcn_wmma_f32_16x16x4_f32)
#define HAVE_WMMA_F32X4 1
#endif

// ---------- wave32 reduction ----------
__device__ __forceinline__ float wave_sum32(float v) {
  v += __shfl_xor(v, 16, 32);
  v += __shfl_xor(v, 8, 32);
  v += __shfl_xor(v, 4, 32);
  v += __shfl_xor(v, 2, 32);
  v += __shfl_xor(v, 1, 32);
  return v;
}
__device__ __forceinline__ float dot4(v4f a, v4f b) {
  return a.x * b.x + a.y * b.y + a.z * b.z + a.w * b.w;
}

__global__ __launch_bounds__(BLK) void hh_rope_wmma_kernel(
    const float* __restrict__ q, const float* __restrict__ k,
    const float* __restrict__ pos, const float* __restrict__ reflg,
    float* __restrict__ out)
{
  // Tiles: 16 rows x 132 floats (padded) x 2 buffers x 2 waves = 33 KB
  __shared__ __align__(16) float tile[WPB][PIPE][16][PADD];
  // Zero-padded B matrix (16 cols x 128, padded): rows 0/1 = reflectors of the
  // current head, rows 2..15 permanently zero => unpredicated B-frag loads.
  __shared__ __align__(16) float vbuf[WPB][16][PADD];

  const int lane  = threadIdx.x & 31;
  const int lwave = threadIdx.x >> 5;
  const int gwave = blockIdx.x * WPB + lwave;
  const int nwav  = gridDim.x * WPB;
  if (gwave >= NG) return;
  const int Tw = (NG - 1 - gwave) / nwav + 1;     // groups for this wave

  // One-time zero of B rows 2..15 (wave-private; DS is in-order per wave).
  {
    float* zb = &vbuf[lwave][R_][0];
    for (int i = lane; i < (16 - R_) * PADD; i += 32) zb[i] = 0.0f;
  }

  // RoPE frequency factors: lane l owns pair indices 2l and 2l+1. Hoisted.
  const float f0 = exp2f((float)(2 * lane)     * (-LOG2_BASE / 64.0f));
  const float f1 = exp2f((float)(2 * lane + 1) * (-LOG2_BASE / 64.0f));

  // Stage one 16-row tile (rows strided by H*D in memory; 512B each) via the
  // CDNA5 async engine: 16 x global_load_async_to_lds_b128 (one row per issue).
  auto issue_tile = [&](int g, int slot_) {
    const int t  = (g >= GPT) ? 1 : 0;
    const int gi = g - t * GPT;
    const int h  = gi % H_;
    const int sb = (gi / H_) % SB;
    const int b  = gi / (H_ * SB);
    const float* src = (t ? k : q) +
        ((((long long)b * S_ + (long long)sb * 16) * H_ + h) * D_);
#pragma unroll
    for (int m = 0; m < 16; ++m)
      async_b128(src + (long long)m * ROWSTRIDE + lane * 4,
                 &tile[lwave][slot_][m][lane * 4]);
  };

  const int npre = (Tw < PIPE) ? Tw : PIPE;
  for (int j = 0; j < npre; ++j) issue_tile(gwave + j * nwav, j);

  for (int j = 0; j < Tw; ++j) {
    const int g  = gwave + j * nwav;
    const int t  = (g >= GPT) ? 1 : 0;
    const int gi = g - t * GPT;
    const int h  = gi % H_;
    const int sb = (gi / H_) % SB;
    const int b  = gi / (H_ * SB);
    const long long ebase =
        (((long long)b * S_ + (long long)sb * 16) * H_ + h) * D_;
    const long long obase = (t ? (long long)BSH * D_ : 0) + ebase;

    // Reflectors for this head: registers (update pass) + B rows 0/1 in LDS.
    v4f v0 = ((const v4f*)(reflg + (h * R_ + 0) * D_))[lane];
    v4f v1 = ((const v4f*)(reflg + (h * R_ + 1) * D_))[lane];
    ((v4f*)&vbuf[lwave][0][0])[lane] = v0;
    ((v4f*)&vbuf[lwave][1][0])[lane] = v1;
    const float sq0   = wave_sum32(dot4(v0, v0)) + EPS_;
    const float sq1   = wave_sum32(dot4(v1, v1)) + EPS_;
    const float cross = wave_sum32(dot4(v0, v1));   // v0.v1 for cross-term fix

    // Oldest of the two in-flight tiles (16 issues) is ours in steady state.
    if (j + PIPE <= Tw) wait_asynccnt<16>();
    else                wait_asynccnt<0>();

    const int slot = j & (PIPE - 1);
    const float* tl = &tile[lwave][slot][0][0];

#if defined(HAVE_WMMA_F32X4)
    // ---- Batched dots on the matrix pipe: D = X(16x128) * V^T(128x16) ----
    // A frag (ISA 7.12.2): M = lane%16, K = 2*(lane/16) + vgpr.
    // B frag: N = lane%16 (rows 2..15 of vbuf are zero), K as above.
    // 132-float row pitch => every chunk's b64 reads hit 64 distinct banks.
    const int ml = lane & 15, half = lane >> 4;
    const float* ap = tl + ml * PADD + 2 * half;
    const float* bp = &vbuf[lwave][ml][2 * half];
    v8f acc = {};
#pragma unroll
    for (int kb = 0; kb < 32; ++kb) {
      v2f a  = *(const v2f*)(ap + kb * 4);
      v2f bf = *(const v2f*)(bp + kb * 4);
      acc = __builtin_amdgcn_wmma_f32_16x16x4_f32(
          false, a, false, bf, (short)0, acc, false, false);
    }
#endif

    // ---- Per-row coefficient extraction + update + RoPE + NT store ----
#pragma unroll
    for (int m = 0; m < 16; ++m) {
      v4f x = ((const v4f*)(tl + m * PADD))[lane];
#if defined(HAVE_WMMA_F32X4)
      // D[m][n] lives in VGPR m%8 at lane (m>=8)*16 + n.
      const float d0 = __shfl(acc[m & 7], ((m >> 3) << 4) | 0, 32);
      const float d1 = __shfl(acc[m & 7], ((m >> 3) << 4) | 1, 32);
      const float c0 = 2.0f * d0 / sq0;
      const float c1 = 2.0f * (d1 - c0 * cross) / sq1;   // sequential-equivalent
      x.x -= c0 * v0.x + c1 * v1.x;
      x.y -= c0 * v0.y + c1 * v1.y;
      x.z -= c0 * v0.z + c1 * v1.z;
      x.w -= c0 * v0.w + c1 * v1.w;
#else
      const float c0 = 2.0f * wave_sum32(dot4(x, v0)) / sq0;
      x.x -= c0 * v0.x; x.y -= c0 * v0.y; x.z -= c0 * v0.z; x.w -= c0 * v0.w;
      const float c1 = 2.0f * wave_sum32(dot4(x, v1)) / sq1;
      x.x -= c1 * v1.x; x.y -= c1 * v1.y; x.z -= c1 * v1.z; x.w -= c1 * v1.w;
#endif
      const float p = pos[sb * 16 + m];
      float s0, cc0, s1, cc1;
      __sincosf(p * f0, &s0, &cc0);
      __sincosf(p * f1, &s1, &cc1);
      v4f y;
      y.x = x.x * cc0 - x.y * s0;
      y.y = x.x * s0 + x.y * cc0;
      y.z = x.z * cc1 - x.w * s1;
      y.w = x.z * s1 + x.w * cc1;
      __builtin_nontemporal_store(
          y, (v4f*)(out + obase + (long long)m * ROWSTRIDE) + lane);
    }

    // Refill freed slot (close DS-read vs async-LDS-write WAR, then issue).
    if (j + PIPE < Tw) {
      wait_dscnt0();
      issue_tile(gwave + (j + PIPE) * nwav, slot);
    }
  }
}

extern "C" void kernel_launch(void* const* d_in, const int* in_sizes, int n_in,
                              void* d_out, int out_size, void* d_ws, size_t ws_size,
                              hipStream_t stream) {
  const float* q    = (const float*)d_in[0];
  const float* k    = (const float*)d_in[1];
  const float* pos  = (const float*)d_in[2];
  const float* refl = (const float*)d_in[3];
  float*       out  = (float*)d_out;

  // 2048 blocks x 2 waves = 4096 waves; 8 groups (128 rows) per wave.
  hh_rope_wmma_kernel<<<dim3(2048), dim3(BLK), 0, stream>>>(q, k, pos, refl, out);
  (void)in_sizes; (void)n_in; (void)out_size; (void)d_ws; (void)ws_size;
}